// DeoLayer_58961311039781
// MI455X (gfx1250) — compile-verified
//
#include <hip/hip_runtime.h>
#include <cstdint>
#include <cstddef>

// Problem geometry (fixed by the reference).
#define BATCH   8
#define CH      256
#define GROUPS  8
#define CPG     (CH / GROUPS)   // 32 channels share one offset group
#define OSIZE   8
#define HH      64
#define WW      64

#define WT      32              // w-tile per block
#define WSTRIDE 68              // padded LDS weight row stride (dwords); 272B, 16B aligned

// LDS float layout:
//   [0, 8192)            : x tile, TDM-packed as ((c_local*8 + k)*32 + w)
//   [8192, 8448)         : offsets (8 o x 32 w)
//   [8448, 8448+32*68)   : weights w[w][o*8+k], padded rows
#define XOFF    0
#define OFFOFF  (CPG * OSIZE * WT)          // 8192
#define WGTOFF  (OFFOFF + OSIZE * WT)       // 8448
#define SMEM_FLOATS (WGTOFF + WT * WSTRIDE) // 10624 floats = 42496 B

typedef unsigned int u32x4 __attribute__((ext_vector_type(4)));
typedef unsigned int u32x8 __attribute__((ext_vector_type(8)));

__global__ __launch_bounds__(256)
void deo_orient_sample_kernel(const float* __restrict__ x,
                              const float* __restrict__ off,
                              float* __restrict__ out) {
    __shared__ __align__(128) float smem[SMEM_FLOATS];

    const int tid = threadIdx.x;
    const int b  = blockIdx.z;
    const int g  = blockIdx.y;
    const int h  = blockIdx.x >> 1;
    const int w0 = (blockIdx.x & 1) * WT;

    const unsigned lds_base = (unsigned)(uintptr_t)smem;  // LDS byte offset in low 32 bits

    if (tid < 32) {
        // ---- TDM: one DMA for the whole 3D x tile (32 w, 8 k, 32 c) = 32 KB -> LDS.
        // tile element (x=w, y=k, z=c): global strides 1 / 4096 / 32768 floats.
        {
            const unsigned long long ga = (unsigned long long)(uintptr_t)
                (x + ((((size_t)b * CH + g * CPG) * OSIZE) * HH + h) * WW + w0);
            const u32x4 d0 = {
                1u,                                              // count=1, no gather
                lds_base + XOFF * 4u,                            // lds_addr (bytes)
                (unsigned)ga,                                    // global_addr[31:0]
                (unsigned)((ga >> 32) & 0x1FFFFFFu) | (2u << 30) // addr[56:32] | type=2
            };
            const u32x8 d1 = {
                2u << 16,                 // wg_mask=0, data_size=2 (4B), no flags
                (unsigned)WT << 16,       // tensor_dim0 = 32 (lo16 in [63:48])
                (unsigned)OSIZE << 16,    // tensor_dim1 = 8  (lo16 in [111:96+?] -> dw2[31:16])
                (unsigned)WT << 16,       // tile_dim0 = 32   ([127:112])
                (unsigned)OSIZE | ((unsigned)CPG << 16),  // tile_dim1=8, tile_dim2=32
                (unsigned)(HH * WW),      // tensor_dim0_stride = 4096 (lo32)
                (unsigned)(OSIZE * HH * WW & 0xFFFF) << 16, // dim1_stride lo16 ([223:208])
                (unsigned)((OSIZE * HH * WW) >> 16)         // dim1_stride bits [47:16]
            };
            const u32x4 d2 = { (unsigned)CPG, 0u, 0u, 0u };     // tensor_dim2 = 32
            const u32x4 d3 = { 0u, 0u, 0u, 0u };
            asm volatile("tensor_load_to_lds %0, %1, %2, %3"
                         :: "s"(d0), "s"(d1), "s"(d2), "s"(d3)
                         : "memory");
        }

        // ---- ASYNC: 8x32 offset tile (shared by the 32 channels) -> LDS.
        {
            unsigned lds_a = lds_base + (OFFOFF + tid) * 4u;
            unsigned long long ga = (unsigned long long)(uintptr_t)
                (off + ((((size_t)b * GROUPS + g) * OSIZE) * HH + h) * WW + w0 + tid);
#pragma unroll
            for (int o = 0; o < OSIZE; ++o) {
                asm volatile("global_load_async_to_lds_b32 %0, %1, off"
                             :: "v"(lds_a), "v"(ga)
                             : "memory");
                lds_a += WT * 4u;
                ga    += (unsigned long long)HH * WW * sizeof(float);
            }
            asm volatile("s_wait_asynccnt 0" ::: "memory");
        }
    }
    __syncthreads();   // offsets visible; x-tile DMA still in flight (overlaps phase 1)

    // -------- Phase 1: per-position 8x8 cyclic interpolation matrices.
    // weight(w,o,k) = max(0, 1 - |wrap8(o + offset - k)|) == {1-frac at i0, frac at i1, else 0}
    {
        const int w = tid & (WT - 1);
        const int o = tid >> 5;                       // 256 threads = 32 w x 8 o
        const float p = (float)o + smem[OFFOFF + o * WT + w];
        float* wrow = &smem[WGTOFF + w * WSTRIDE + o * OSIZE];
#pragma unroll
        for (int k = 0; k < OSIZE; ++k) {
            float r = p - (float)k;
            r -= 8.0f * floorf(fmaf(r, 0.125f, 0.5f));   // wrap to [-4, 4)
            wrow[k] = fmaxf(0.0f, 1.0f - fabsf(r));
        }
    }
    __builtin_amdgcn_s_wait_tensorcnt(0);   // issuing wave drains TDM; others: no-op
    __syncthreads();                        // x tile + weights now visible to all

    // -------- Phase 2: stream from LDS, 8 FMAs per output element, NT stores.
    const int w   = tid & (WT - 1);
    const int ch0 = tid >> 5;                         // 0..7

    float wt[OSIZE * OSIZE];
    {
        const float* wbase = &smem[WGTOFF + w * WSTRIDE];
#pragma unroll
        for (int i = 0; i < OSIZE * OSIZE; ++i) wt[i] = wbase[i];   // ds_load_b128 x16
    }

#pragma unroll
    for (int cs = 0; cs < 4; ++cs) {
        const int cl = ch0 + 8 * cs;                  // channel within group, 0..31
        const float* xrow = &smem[XOFF + cl * (OSIZE * WT) + w];
        float xv[OSIZE];
#pragma unroll
        for (int k = 0; k < OSIZE; ++k)
            xv[k] = xrow[k * WT];                     // conflict-free ds_load_b32

        const size_t base =
            ((((size_t)b * CH + g * CPG + cl) * OSIZE) * HH + h) * WW + w0 + w;
#pragma unroll
        for (int o = 0; o < OSIZE; ++o) {
            float acc = 0.0f;
#pragma unroll
            for (int k = 0; k < OSIZE; ++k)
                acc = fmaf(xv[k], wt[o * OSIZE + k], acc);
            __builtin_nontemporal_store(acc, out + base + (size_t)o * HH * WW);
        }
    }
}

extern "C" void kernel_launch(void* const* d_in, const int* in_sizes, int n_in,
                              void* d_out, int out_size, void* d_ws, size_t ws_size,
                              hipStream_t stream) {
    (void)in_sizes; (void)n_in; (void)out_size; (void)d_ws; (void)ws_size;
    const float* x   = (const float*)d_in[0];
    const float* off = (const float*)d_in[1];
    float* out       = (float*)d_out;

    dim3 grid(HH * (WW / WT), GROUPS, BATCH);   // (128, 8, 8) = 8192 blocks
    deo_orient_sample_kernel<<<grid, 256, 0, stream>>>(x, off, out);
}